// KNNEntropyEstimator_47880295415991
// MI455X (gfx1250) — compile-verified
//
#include <hip/hip_runtime.h>

// KNN (Kozachenko-Leonenko style) entropy estimator, N=2048, D=16, k=5.
//
// Math collapse (exact, tie-safe): kth-smallest signed diff per (i,c) equals
// x[i,c] - m_c where m_c is the (k+1)=6th largest value of column c. Then
//   d = x - m_c; hi = min(x + d, 1); lo = max(x - d, 0); eps = hi - lo
// is bit-identical to the reference's per-element arithmetic, and
//   H = (digamma(16) - digamma(5) + 3) + mean_i sum_c eps.
//
// Single workgroup (512 threads = 16 wave32s on one WGP):
//   Phase 1: wave w selects the 6th largest of column w (lane-local streaming
//            top-6, then a 192-candidate merge by lane 0).
//   Phase 2: all waves fold the 32768 eps values through
//            V_WMMA_F32_16X16X4_F32 with an all-ones B matrix (full fp32
//            accumulation), then a wave32 shuffle reduction.

#define N_SAMP 2048
#define DIMS   16

// digamma(16) - digamma(5) + (16-1)/5 = sum_{n=5}^{15} 1/n + 3  (double prec.)
#define H_CONST 4.2348956598956598f

typedef float v2f __attribute__((ext_vector_type(2)));
typedef float v8f __attribute__((ext_vector_type(8)));

// Insert v into a descending sorted 6-list held in registers.
#define TOP6_INSERT(v, t0, t1, t2, t3, t4, t5)                                 \
  do {                                                                         \
    float _v = (v);                                                            \
    if (_v > t5) {                                                             \
      if (_v > t0)      { t5=t4; t4=t3; t3=t2; t2=t1; t1=t0; t0=_v; }          \
      else if (_v > t1) { t5=t4; t4=t3; t3=t2; t2=t1; t1=_v; }                 \
      else if (_v > t2) { t5=t4; t4=t3; t3=t2; t2=_v; }                        \
      else if (_v > t3) { t5=t4; t4=t3; t3=_v; }                               \
      else if (_v > t4) { t5=t4; t4=_v; }                                      \
      else              { t5=_v; }                                             \
    }                                                                          \
  } while (0)

__global__ __launch_bounds__(512)
void knn_entropy_wmma(const float* __restrict__ x, float* __restrict__ out)
{
  __shared__ float cand[DIMS][6 * 32];  // per-column candidates (lane top-6s)
  __shared__ float mcol[DIMS];          // 6th-largest per column
  __shared__ float wsum[DIMS];          // per-wave partial sums

  const int tid  = (int)threadIdx.x;
  const int wave = tid >> 5;            // 0..15  (wave32!)
  const int lane = tid & 31;

  // ---------------- Phase 1: per-column 6th-largest ----------------
  {
    const int c = wave;                 // one wave per column
    float t0 = -3.0e38f, t1 = -3.0e38f, t2 = -3.0e38f,
          t3 = -3.0e38f, t4 = -3.0e38f, t5 = -3.0e38f;
    for (int j = lane; j < N_SAMP; j += 32) {
      float v = x[j * DIMS + c];
      TOP6_INSERT(v, t0, t1, t2, t3, t4, t5);
    }
    float* cc = &cand[c][lane * 6];
    cc[0] = t0; cc[1] = t1; cc[2] = t2; cc[3] = t3; cc[4] = t4; cc[5] = t5;
  }
  __syncthreads();

  if (lane == 0) {
    const int c = wave;
    float t0 = -3.0e38f, t1 = -3.0e38f, t2 = -3.0e38f,
          t3 = -3.0e38f, t4 = -3.0e38f, t5 = -3.0e38f;
    for (int q = 0; q < 6 * 32; ++q) {
      float v = cand[c][q];
      TOP6_INSERT(v, t0, t1, t2, t3, t4, t5);
    }
    mcol[c] = t5;                       // the 6th largest of column c
  }
  __syncthreads();

  // ---------------- Phase 2: WMMA fp32 accumulation of eps ----------------
  // Wave w owns flat elements [w*2048, (w+1)*2048). Lane l, iter t handles
  // elements e = w*2048 + t*64 + 2*l and e+1. Since 2048 and 64 are multiples
  // of 16, the column of e is (2*l) & 15 — constant per lane.
  const int   col0 = (2 * lane) & 15;
  const float m0   = mcol[col0];
  const float m1   = mcol[col0 + 1];

  v8f acc = {};                         // 16x16 f32 accumulator (8 VGPRs)
  v2f bones; bones[0] = 1.0f; bones[1] = 1.0f;   // B = ones(4x16)

  const float2* xv2  = (const float2*)x;
  const int     base = wave * ((N_SAMP * DIMS / 16) / 2) + lane;  // float2 units

  for (int t = 0; t < 32; ++t) {
    float2 v = xv2[base + t * 32];

    float d0 = v.x - m0;
    float h0 = fminf(v.x + d0, 1.0f);
    float l0 = fmaxf(v.x - d0, 0.0f);

    float d1 = v.y - m1;
    float h1 = fminf(v.y + d1, 1.0f);
    float l1 = fmaxf(v.y - d1, 0.0f);

    v2f a; a[0] = h0 - l0; a[1] = h1 - l1;   // A = 16x4 chunk of eps values

    // D = A x ones + C : every A element is summed into its row of C.
    // 8 args: (neg_a, A, neg_b, B, c_mod, C, reuse_a, reuse_b)
    acc = __builtin_amdgcn_wmma_f32_16x16x4_f32(
        false, a, false, bones, (short)0, acc, false, false);
  }

  // Each of the 16 C-columns replicates the row sums, so the true wave total
  // is (sum of all lanes' 8 accumulator VGPRs) / 16.
  float part = 0.0f;
  for (int q = 0; q < 8; ++q) part += acc[q];
  for (int off = 16; off > 0; off >>= 1)
    part += __shfl_xor(part, off, 32);        // wave32 butterfly
  if (lane == 0) wsum[wave] = part * (1.0f / 16.0f);
  __syncthreads();

  if (tid == 0) {
    float S = 0.0f;
    for (int w = 0; w < DIMS; ++w) S += wsum[w];
    out[0] = H_CONST + S * (1.0f / (float)N_SAMP);
  }
}

extern "C" void kernel_launch(void* const* d_in, const int* in_sizes, int n_in,
                              void* d_out, int out_size, void* d_ws, size_t ws_size,
                              hipStream_t stream) {
  (void)in_sizes; (void)n_in; (void)out_size; (void)d_ws; (void)ws_size;
  const float* x = (const float*)d_in[0];
  // d_in[1] is k (==5); k and the digamma terms are folded into H_CONST.
  knn_entropy_wmma<<<1, 512, 0, stream>>>(x, (float*)d_out);
}